// NGPRadianceField_35673998360638
// MI455X (gfx1250) — compile-verified
//
#include <hip/hip_runtime.h>
#include <stdint.h>

// ---------------------------------------------------------------------------
// Instant-NGP multiresolution hash-grid encoding, forward pass (MI455X).
// 1,048,576 points x 16 levels x 2 features. Hash table: 16 * 2^19 float2
// rows = 64 MB -> resident in the 192 MB L2.
//
// Roofline: ~0.6 GFLOP (irrelevant, no WMMA structure); 128 MB streamed
// output (~6 us @ 23.3 TB/s); 1 GB of logical b64 gathers served from L2.
// => L2 gather-request-rate bound. Strategy:
//  - thread = (point, level): 8 independent global_load_b64 gathers in flight
//  - NT store for write-once output (don't evict the table from L2)
//  - b128-staged point coords through LDS (12 loads / 256-thread block)
//  - per-level scale floor(2^(4+8l/15)) via raw v_exp_f32 (arg in [4,12],
//    no libm denormal fixup needed)
//  - guard-free hot kernel; guarded tail kernel only if n % 16 != 0
// ---------------------------------------------------------------------------

typedef float v2f __attribute__((ext_vector_type(2)));
typedef float v4f __attribute__((ext_vector_type(4)));

#define NGP_T  (1u << 19)          // hash table size per level
#define NGP_P1 2654435761u
#define NGP_P2 805459861u

template<bool GUARD>
__global__ __launch_bounds__(256) void ngp_hashgrid_fwd(
    const float* __restrict__ pts,     // [n,3] fp32
    const float* __restrict__ table,   // [16*2^19, 2] fp32
    float*       __restrict__ out,     // [n,32] fp32
    int n_points)
{
    // 256 threads = 16 points x 16 levels per block.
    __shared__ v4f sp4[12];            // 16 points * 3 floats = 12 x b128
    float* sp = (float*)sp4;
    const int tid = threadIdx.x;
    const int p0  = blockIdx.x << 4;

    if (!GUARD) {
        // Full block: 12 aligned b128 loads stage 16 points into LDS.
        if (tid < 12)
            sp4[tid] = ((const v4f*)(pts + (size_t)p0 * 3))[tid];
    } else {
        if (tid < 48) {
            const int g = p0 * 3 + tid;
            sp[tid] = (g < n_points * 3) ? pts[g] : 0.0f;
        }
    }
    __syncthreads();

    const int lp  = tid >> 4;          // local point  0..15
    const int lvl = tid & 15;          // level        0..15
    const int p   = p0 + lp;
    if (GUARD && p >= n_points) return;

    // Per-level resolution: floor(16 * (256^(1/15))^lvl) = floor(2^(4+8l/15)).
    // Argument is in [4, 12] -> raw hardware v_exp_f32 is safe (no denormal
    // input fixup required), and endpoints (16, 4096) are exact.
    const float s  = floorf(__builtin_amdgcn_exp2f(
                         fmaf((float)lvl, 8.0f / 15.0f, 4.0f)));
    const float sx = sp[lp * 3 + 0] * s;
    const float sy = sp[lp * 3 + 1] * s;
    const float sz = sp[lp * 3 + 2] * s;

    const float fxf = floorf(sx), fyf = floorf(sy), fzf = floorf(sz);
    const float ox = sx - fxf, oy = sy - fyf, oz = sz - fzf;

    const uint32_t fx = (uint32_t)(int)fxf;
    const uint32_t fy = (uint32_t)(int)fyf;
    const uint32_t fz = (uint32_t)(int)fzf;
    const uint32_t cx = (uint32_t)(int)ceilf(sx);
    const uint32_t cy = (uint32_t)(int)ceilf(sy);
    const uint32_t cz = (uint32_t)(int)ceilf(sz);

    // Hash: x*1 ^ y*P1 ^ z*P2, masked to T-1 (T is a power of two).
    const uint32_t hyf = fy * NGP_P1, hyc = cy * NGP_P1;
    const uint32_t hzf = fz * NGP_P2, hzc = cz * NGP_P2;
    const uint32_t mask = NGP_T - 1u;

    const v2f* __restrict__ tab = (const v2f*)table + (uint32_t)lvl * NGP_T;

    // 8 independent b64 gathers — all issued before first use.
    const v2f f0 = tab[(cx ^ hyc ^ hzc) & mask];   // (cx,cy,cz)
    const v2f f1 = tab[(cx ^ hyf ^ hzc) & mask];   // (cx,fy,cz)
    const v2f f2 = tab[(fx ^ hyf ^ hzc) & mask];   // (fx,fy,cz)
    const v2f f3 = tab[(fx ^ hyc ^ hzc) & mask];   // (fx,cy,cz)
    const v2f f4 = tab[(cx ^ hyc ^ hzf) & mask];   // (cx,cy,fz)
    const v2f f5 = tab[(cx ^ hyf ^ hzf) & mask];   // (cx,fy,fz)
    const v2f f6 = tab[(fx ^ hyf ^ hzf) & mask];   // (fx,fy,fz)
    const v2f f7 = tab[(fx ^ hyc ^ hzf) & mask];   // (fx,cy,fz)

    // Trilinear interpolation; a*w + b*(1-w) == b + w*(a-b) -> packed FMA.
    const v2f f03 = f3 + ox * (f0 - f3);
    const v2f f12 = f2 + ox * (f1 - f2);
    const v2f f56 = f6 + ox * (f5 - f6);
    const v2f f47 = f7 + ox * (f4 - f7);
    const v2f fa  = f12 + oy * (f03 - f12);
    const v2f fb  = f56 + oy * (f47 - f56);
    const v2f enc = fb  + oz * (fa  - fb);

    // Non-temporal streaming store (TH=NT): keeps the 64 MB hash table hot in
    // L2. The 16 lanes of one point write one contiguous 128 B segment.
    __builtin_nontemporal_store(enc, (v2f*)out + (size_t)p * 16 + lvl);
}

extern "C" void kernel_launch(void* const* d_in, const int* in_sizes, int n_in,
                              void* d_out, int out_size, void* d_ws, size_t ws_size,
                              hipStream_t stream)
{
    const float* pts   = (const float*)d_in[0];   // in_tensor  [n,3]
    const float* table = (const float*)d_in[1];   // hash_table [16*2^19,2]
    float*       out   = (float*)d_out;           // [n,32]

    const int n    = in_sizes[0] / 3;
    const int full = n >> 4;                      // full 16-point blocks
    const int rem  = n & 15;

    if (full > 0) {
        hipLaunchKernelGGL((ngp_hashgrid_fwd<false>), dim3(full), dim3(256), 0,
                           stream, pts, table, out, n);
    }
    if (rem > 0) {
        // Guarded tail: one block over the last (n % 16) points.
        hipLaunchKernelGGL((ngp_hashgrid_fwd<true>), dim3(1), dim3(256), 0,
                           stream,
                           pts + (size_t)full * 16 * 3,
                           table,
                           out + (size_t)full * 16 * 32,
                           rem);
    }
}